// E2M1ActivationQuantizer_635655160011
// MI455X (gfx1250) — compile-verified
//
#include <hip/hip_runtime.h>
#include <hip/hip_bf16.h>

// ---------------------------------------------------------------------------
// MX-FP4 (E2M1, block=32) quantize + dequantize for MI455X (gfx1250).
// Each lane owns one 32-element quant block (8x b128 NT loads). Encode packs
// 4-bit codes via a ties-down bit-trick; dequant uses the gfx1250-native
// v_cvt_scale_pk8_f32_fp4 (8 elements per instruction, E8M0 scale applied in
// hardware). Known-present from round-4 probe: cvt_scale_pk8_{f32,f16,bf16}_fp4,
// cvt_scale_pk8_f32_fp8, cvt_scale_pk16_f32_fp6, global_load_async_to_lds_b128.
//
// PROBE wave 2 (encode-direction candidates), result encoded in kernel name:
//  Q0 cvt_scale_sr_pk8_fp4_bf16   Q1 cvt_sr_pk_fp4_f16
//  Q2 cvt_sr_pk_fp4_f32           Q3 cvt_pk_fp4_f16
//  Q4 cvt_pk_fp4_bf16             Q5 cvt_scale_sr_pk16_fp4_f32
//  Q6 cvt_scale_pk8_fp4_f32       Q7 cvt_scale_sr_pk_fp4_f32
// ---------------------------------------------------------------------------

#define PCAT2(a, b) a##b
#define PCAT(a, b) PCAT2(a, b)

#if __has_builtin(__builtin_amdgcn_cvt_scale_sr_pk8_fp4_bf16)
#define QB0 1
#else
#define QB0 0
#endif
#if __has_builtin(__builtin_amdgcn_cvt_sr_pk_fp4_f16)
#define QB1 1
#else
#define QB1 0
#endif
#if __has_builtin(__builtin_amdgcn_cvt_sr_pk_fp4_f32)
#define QB2 1
#else
#define QB2 0
#endif
#if __has_builtin(__builtin_amdgcn_cvt_pk_fp4_f16)
#define QB3 1
#else
#define QB3 0
#endif
#if __has_builtin(__builtin_amdgcn_cvt_pk_fp4_bf16)
#define QB4 1
#else
#define QB4 0
#endif
#if __has_builtin(__builtin_amdgcn_cvt_scale_sr_pk16_fp4_f32)
#define QB5 1
#else
#define QB5 0
#endif
#if __has_builtin(__builtin_amdgcn_cvt_scale_pk8_fp4_f32)
#define QB6 1
#else
#define QB6 0
#endif
#if __has_builtin(__builtin_amdgcn_cvt_scale_sr_pk_fp4_f32)
#define QB7 1
#else
#define QB7 0
#endif

#define QN0 PCAT(e2m1_hw_, QB0)
#define QN1 PCAT(QN0, QB1)
#define QN2 PCAT(QN1, QB2)
#define QN3 PCAT(QN2, QB3)
#define QN4 PCAT(QN3, QB4)
#define QN5 PCAT(QN4, QB5)
#define QN6 PCAT(QN5, QB6)
#define KNAME PCAT(QN6, QB7)

typedef float v4f __attribute__((ext_vector_type(4)));
typedef float v8f __attribute__((ext_vector_type(8)));

// 4-bit E2M1 code (sign<<3 | ord) of x/scale, ties-down (== reference's
// strict-> compares against midpoints {0.25,0.75,1.25,1.75,2.5,3.5,5}).
__device__ __forceinline__ unsigned int code_one(float x, float rscale) {
  float a = fabsf(x) * rscale;              // exact: rscale is a power of two
  float c = fminf(a, 6.0f);
  // round to nearest 1-mantissa-bit float, ties toward zero:
  unsigned int r_u = (__float_as_uint(c) + 0x1FFFFFu) & 0xFFC00000u;
  int ord_hi = (int)(r_u >> 22) - 252;      // valid when a > 0.875 (-> >= 1.0)
  int ord_lo = ((a > 0.25f) ? 1 : 0) + ((a > 0.75f) ? 1 : 0);
  int ord = (a > 0.875f) ? ord_hi : ord_lo;
  return (unsigned int)(ord | ((x > 0.0f) ? 0 : 8));
}

__global__ void __launch_bounds__(256)
KNAME(const float* __restrict__ x,
      float* __restrict__ out_deq,
      float* __restrict__ out_enc,
      float* __restrict__ out_scale,
      int n) {
  const long gid  = (long)blockIdx.x * blockDim.x + threadIdx.x;
  const long base = gid * 32;               // one 32-elem block per lane
  if (base >= n) return;

  const v4f* p = reinterpret_cast<const v4f*>(x + base);
  v4f v[8];
#pragma unroll
  for (int j = 0; j < 8; ++j) v[j] = __builtin_nontemporal_load(p + j);

  // ---- block amax: in-lane v_max3-foldable chains, then a combine tree ---
  float mv[8];
#pragma unroll
  for (int j = 0; j < 8; ++j) {
    mv[j] = fmaxf(fmaxf(fmaxf(fabsf(v[j].x), fabsf(v[j].y)), fabsf(v[j].z)),
                  fabsf(v[j].w));
  }
  float m03 = fmaxf(fmaxf(fmaxf(mv[0], mv[1]), mv[2]), mv[3]);
  float m47 = fmaxf(fmaxf(fmaxf(mv[4], mv[5]), mv[6]), mv[7]);
  float amax = fmaxf(m03, m47);

  // ---- e = ceil(log2(amax/6)) clamped at -127, from the raw bits ---------
  unsigned int u = __float_as_uint(amax);
  int E = (int)(u >> 23);
  int e = (E - 129) + (((u & 0x7FFFFFu) > 0x400000u) ? 1 : 0);
  e = (e < -127) ? -127 : e;
  unsigned int scale_u8 = (unsigned int)(e + 127);   // E8M0 biased exponent
  float rscale = ldexpf(1.0f, -e);                   // exact, v_ldexp_f32

  out_scale[gid] = (float)scale_u8;         // coalesced b32, one per block

#pragma unroll
  for (int j2 = 0; j2 < 4; ++j2) {
    v4f A = v[2 * j2], B = v[2 * j2 + 1];
    float w[8] = {A.x, A.y, A.z, A.w, B.x, B.y, B.z, B.w};

    // ---- encode: 8x 4-bit codes, packed into one dword ----
    unsigned int c8 = 0;
    float encf[8];
#pragma unroll
    for (int k = 0; k < 8; ++k) {
      unsigned int code = code_one(w[k], rscale);
      encf[k] = (float)code;                // one v_cvt_f32_u32 per element
      c8 |= code << (4 * k);
    }

    // ---- dequant: gfx1250 v_cvt_scale_pk8_f32_fp4 -------------------------
    // 8 fp4 codes -> 8 f32, multiplied by 2^(scale_u8-127) in hardware.
    v8f d = __builtin_amdgcn_cvt_scale_pk8_f32_fp4(c8, scale_u8, 0);

    v4f dlo = {d.s0, d.s1, d.s2, d.s3};
    v4f dhi = {d.s4, d.s5, d.s6, d.s7};
    v4f elo = {encf[0], encf[1], encf[2], encf[3]};
    v4f ehi = {encf[4], encf[5], encf[6], encf[7]};
    __builtin_nontemporal_store(dlo, reinterpret_cast<v4f*>(out_deq + base) + 2 * j2);
    __builtin_nontemporal_store(dhi, reinterpret_cast<v4f*>(out_deq + base) + 2 * j2 + 1);
    __builtin_nontemporal_store(elo, reinterpret_cast<v4f*>(out_enc + base) + 2 * j2);
    __builtin_nontemporal_store(ehi, reinterpret_cast<v4f*>(out_enc + base) + 2 * j2 + 1);
  }
}

extern "C" void kernel_launch(void* const* d_in, const int* in_sizes, int n_in,
                              void* d_out, int out_size, void* d_ws, size_t ws_size,
                              hipStream_t stream) {
  (void)n_in; (void)out_size; (void)d_ws; (void)ws_size;
  const float* x = (const float*)d_in[0];
  float* out = (float*)d_out;
  const int n = in_sizes[0];                 // 4096*8192 = 33,554,432
  float* out_deq   = out;                    // [0, n)
  float* out_enc   = out + (size_t)n;        // [n, 2n)
  float* out_scale = out + 2 * (size_t)n;    // [2n, 2n + n/32)

  const int threads = n / 32;                // one block per lane
  const int block = 256;                     // 8 waves / workgroup
  const int grid = (threads + block - 1) / block;
  KNAME<<<grid, block, 0, stream>>>(x, out_deq, out_enc, out_scale, n);
}